// HausdorffDistanceLoss_8624294331223
// MI455X (gfx1250) — compile-verified
//
#include <hip/hip_runtime.h>
#include <hip/hip_bf16.h>
#include <math.h>

// ---------------------------------------------------------------------------
// HausdorffDistanceLoss for MI455X (gfx1250, wave32).
//
// Stage 1: boundary extraction (memory-bound, ~8MB traffic) + atomic compaction
//          of boundary coords (reference caps at K=8192, pads with -1).
// Stage 2: pack B operands once, in WMMA lane layout, with the validity mask
//          folded into the norm term (invalid -> 1e30, which dominates d^2).
// Stage 3: pairwise min-distance via V_WMMA_F32_16X16X4_F32:
//          d^2(i,j) = [r, c, r^2+c^2, 1] . [-2r', -2c', 1, norm']  (rank-4).
//          Each wave owns TWO 16-row i-tiles; per j-tile: 1 b64 load -> 2 WMMA
//          -> 8 v_min3. Speculative constant-offset prefetch (folds to imm).
// Stage 4: deterministic tree reduction -> scalar out (with nonempty guard).
// ---------------------------------------------------------------------------

#define KCAP 8192
#define NT   (KCAP / 16)      // 512 j-tiles per side
#define IMH  1024
#define IMW  1024
#define BIGD 1e30f

typedef __attribute__((ext_vector_type(2))) float v2f;
typedef __attribute__((ext_vector_type(8))) float v8f;

// Workspace layout (floats):
//   [0,K)   R1  [K,2K)  C1  [2K,3K) R2  [3K,4K) C2  [4K,5K) V1  [5K,6K) V2
//   [6K, 6K+2*NT)        per-i-tile partial sums (dir0 then dir1)
//   [6K+2*NT, +2)        2 ints: boundary counts (pred, target)
//   [6K+2*NT+2, ...)     packed B operands: 2 sets * NT tiles * 32 lanes * v2f
#define PART_OFF  (6 * KCAP)
#define CNT_OFF   (6 * KCAP + 2 * NT)
#define BOP_OFF   (6 * KCAP + 2 * NT + 2)       // even -> 8-byte aligned
#define BOP_SET   (NT * 32)                      // v2f elements per set

__global__ void hd_init_kernel(float* __restrict__ ws, int* __restrict__ cnt) {
    int i = blockIdx.x * blockDim.x + threadIdx.x;
    if (i < CNT_OFF) ws[i] = (i < 4 * KCAP) ? -1.0f : 0.0f;  // coord pad = -1 (ref fill)
    if (i < 2) cnt[i] = 0;
}

// which==0: apply sigmoid before threshold (pred); which==1: raw (target).
__global__ void hd_boundary_kernel(const float* __restrict__ img, int which,
                                   float* __restrict__ r, float* __restrict__ c,
                                   float* __restrict__ v, int* __restrict__ cnt) {
    int x = blockIdx.x * blockDim.x + threadIdx.x;
    int y = blockIdx.y;
    if (x >= IMW) return;
    float s_dil = 0.0f, s_neg = 0.0f;
    #pragma unroll
    for (int dy = -1; dy <= 1; ++dy) {
        #pragma unroll
        for (int dx = -1; dx <= 1; ++dx) {
            int yy = y + dy, xx = x + dx;
            float b = 0.0f;
            if (yy >= 0 && yy < IMH && xx >= 0 && xx < IMW) {
                float m = img[yy * IMW + xx];
                if (which == 0) m = 1.0f / (1.0f + expf(-m));   // sigmoid
                b = (m > 0.5f) ? 1.0f : 0.0f;
            }
            s_dil += b;       // dilated = conv3x3(b)
            s_neg += -b;      // eroded  = -conv3x3(-b)  (== dilated; faithful)
        }
    }
    float eroded = -s_neg;
    if ((s_dil - eroded) > 0.0f) {
        int idx = atomicAdd(cnt, 1);
        if (idx < KCAP) {
            r[idx] = (float)y;
            c[idx] = (float)x;
            v[idx] = 1.0f;
        }
    }
}

// Pack B operands in WMMA 4x16 f32 lane layout; mask folded into norm term.
// set 0 = target points (consumed by dir0), set 1 = pred points (dir1).
__global__ void hd_pack_kernel(const float* __restrict__ ws, v2f* __restrict__ bop) {
    int t = blockIdx.x * blockDim.x + threadIdx.x;   // 2 * NT * 32 threads
    int set  = t >> 14;                              // NT*32 = 16384
    int rem  = t & (BOP_SET - 1);
    int tile = rem >> 5;
    int lane = rem & 31;
    int half = lane >> 4, lid = lane & 15;
    const float* R = ws + (set ? 0        : 2 * KCAP);
    const float* C = ws + (set ? KCAP     : 3 * KCAP);
    const float* V = ws + (set ? 4 * KCAP : 5 * KCAP);
    int j = tile * 16 + lid;
    float r = R[j], c = C[j], vv = V[j];
    v2f b;
    if (half == 0) { b.x = -2.0f * r; b.y = -2.0f * c; }            // K=0,1 rows
    else           { b.x = 1.0f;      b.y = (vv > 0.5f) ? (r * r + c * c) : BIGD; } // K=2,3
    bop[set * BOP_SET + tile * 32 + lane] = b;
}

// One wave per (direction, pair of i-tiles). 512 waves; blockDim=128 (4 waves).
__global__ void __launch_bounds__(128)
hd_dist_wmma_kernel(const float* __restrict__ ws, const v2f* __restrict__ bop,
                    float* __restrict__ part) {
    int wave = (blockIdx.x * blockDim.x + threadIdx.x) >> 5;
    int dir  = wave >> 8;                 // 256 waves per direction
    int wt   = wave & 255;
    int t0   = wt * 2, t1 = t0 + 1;

    const float* SR1 = ws + (dir ? 2 * KCAP : 0);
    const float* SC1 = ws + (dir ? 3 * KCAP : KCAP);
    const float* SV1 = ws + (dir ? 5 * KCAP : 4 * KCAP);
    const v2f*   B   = bop + dir * BOP_SET;

    int lane = threadIdx.x & 31;
    int half = lane >> 4;
    int lid  = lane & 15;

    // A (16x4 f32): lanes 0-15 row M=lid hold K=0,1; lanes 16-31 hold K=2,3.
    v2f A0, A1;
    {
        float r = SR1[t0 * 16 + lid], c = SC1[t0 * 16 + lid], n = r * r + c * c;
        A0.x = half ? n : r; A0.y = half ? 1.0f : c;
    }
    {
        float r = SR1[t1 * 16 + lid], c = SC1[t1 * 16 + lid], n = r * r + c * c;
        A1.x = half ? n : r; A1.y = half ? 1.0f : c;
    }

    v8f m0, m1;
    #pragma unroll
    for (int v = 0; v < 8; ++v) { m0[v] = BIGD; m1[v] = BIGD; }

    const v2f* bp = B + lane;             // streaming pointer, +32 v2f per tile
    for (int jt = 0; jt < NT; ++jt) {
        v2f Bv = bp[jt * 32];                                  // global_load_b64
        // Speculative prefetch 16 tiles (4KB) ahead at a CONSTANT offset from the
        // load address: folds into the instruction immediate; OOB tail is silently
        // dropped by speculative-prefetch semantics (ISA 10.5).
        __builtin_prefetch((const char*)&bp[jt * 32] + 16 * 32 * sizeof(v2f), 0, 0);
        v8f Cz = {0.f, 0.f, 0.f, 0.f, 0.f, 0.f, 0.f, 0.f};
        v8f D0 = __builtin_amdgcn_wmma_f32_16x16x4_f32(
            false, A0, false, Bv, (short)0, Cz, false, false);
        v8f D1 = __builtin_amdgcn_wmma_f32_16x16x4_f32(
            false, A1, false, Bv, (short)0, Cz, false, false);
        #pragma unroll
        for (int v = 0; v < 8; ++v) {
            m0[v] = fminf(m0[v], D0[v]);               // invalid cols carry 1e30 already
            m1[v] = fminf(m1[v], D1[v]);
        }
    }

    // Min across the 16 columns (lanes) within each half.
    #pragma unroll
    for (int v = 0; v < 8; ++v) {
        float a = m0[v], b = m1[v];
        a = fminf(a, __shfl_xor(a, 1, 32));  b = fminf(b, __shfl_xor(b, 1, 32));
        a = fminf(a, __shfl_xor(a, 2, 32));  b = fminf(b, __shfl_xor(b, 2, 32));
        a = fminf(a, __shfl_xor(a, 4, 32));  b = fminf(b, __shfl_xor(b, 4, 32));
        a = fminf(a, __shfl_xor(a, 8, 32));  b = fminf(b, __shfl_xor(b, 8, 32));
        m0[v] = a; m1[v] = b;
    }

    // Rows 0-7 live in half 0, rows 8-15 in half 1; lane lid==0 of each half sums.
    float s0 = 0.0f, s1 = 0.0f;
    if (lid == 0) {
        #pragma unroll
        for (int v = 0; v < 8; ++v) {
            int mrow = v + 8 * half;
            float q0 = m0[v];
            float d0 = (q0 > 1e29f) ? BIGD : sqrtf(q0);   // no valid target -> BIG (ref)
            if (SV1[t0 * 16 + mrow] > 0.5f) s0 += d0;
            float q1 = m1[v];
            float d1 = (q1 > 1e29f) ? BIGD : sqrtf(q1);
            if (SV1[t1 * 16 + mrow] > 0.5f) s1 += d1;
        }
    }
    s0 += __shfl_xor(s0, 16, 32);
    s1 += __shfl_xor(s1, 16, 32);
    if (lane == 0) {
        part[dir * NT + t0] = s0;
        part[dir * NT + t1] = s1;
    }
}

__global__ void hd_finalize_kernel(const float* __restrict__ part,
                                   const int* __restrict__ cnt,
                                   float* __restrict__ out) {
    __shared__ float a[NT];
    __shared__ float b[NT];
    int t = threadIdx.x;
    a[t] = part[t];
    b[t] = part[NT + t];
    __syncthreads();
    for (int off = NT / 2; off > 0; off >>= 1) {
        if (t < off) { a[t] += a[t + off]; b[t] += b[t + off]; }
        __syncthreads();
    }
    if (t == 0) {
        int cp = cnt[0], ct = cnt[1];
        int n1i = cp < KCAP ? cp : KCAP; if (n1i < 1) n1i = 1;   // n1 = max(sum(v1),1)
        int n2i = ct < KCAP ? ct : KCAP; if (n2i < 1) n2i = 1;
        float d_p2t = a[0] / (float)n1i;
        float d_t2p = b[0] / (float)n2i;
        float hd = fmaxf(d_p2t, d_t2p);
        out[0] = (cp > 0 && ct > 0) ? hd : 0.0f;                 // nonempty guard
    }
}

extern "C" void kernel_launch(void* const* d_in, const int* in_sizes, int n_in,
                              void* d_out, int out_size, void* d_ws, size_t ws_size,
                              hipStream_t stream) {
    const float* pred   = (const float*)d_in[0];
    const float* target = (const float*)d_in[1];
    float* ws   = (float*)d_ws;
    float* part = ws + PART_OFF;
    int*   cnt  = (int*)(ws + CNT_OFF);
    v2f*   bop  = (v2f*)(ws + BOP_OFF);

    hd_init_kernel<<<(CNT_OFF + 255) / 256, 256, 0, stream>>>(ws, cnt);

    dim3 bgrid((IMW + 255) / 256, IMH);
    hd_boundary_kernel<<<bgrid, 256, 0, stream>>>(pred,   0, ws,            ws + KCAP,     ws + 4 * KCAP, cnt + 0);
    hd_boundary_kernel<<<bgrid, 256, 0, stream>>>(target, 1, ws + 2 * KCAP, ws + 3 * KCAP, ws + 5 * KCAP, cnt + 1);

    // Pack B operands: 2 sets * 512 tiles * 32 lanes = 32768 threads.
    hd_pack_kernel<<<(2 * BOP_SET) / 256, 256, 0, stream>>>(ws, bop);

    // 2 directions * 256 tile-pairs = 512 waves; 4 waves/block -> 128 blocks.
    hd_dist_wmma_kernel<<<128, 128, 0, stream>>>(ws, bop, part);

    hd_finalize_kernel<<<1, NT, 0, stream>>>(part, cnt, (float*)d_out);
}